// LocalPODLoss_48996986913141
// MI455X (gfx1250) — compile-verified
//
#include <hip/hip_runtime.h>
#include <math.h>
#include <stdint.h>

typedef __attribute__((ext_vector_type(2))) float v2f;
typedef __attribute__((ext_vector_type(8))) float v8f;

#define NMAPS 8192                   // B*C = 32*256
#define MPW 16                       // maps per workgroup
#define NBLK (NMAPS / MPW)           // 512 workgroups per scale

#if __has_builtin(__builtin_amdgcn_s_wait_asynccnt)
#define WAIT_ASYNC(n) __builtin_amdgcn_s_wait_asynccnt(n)
#else
#define WAIT_ASYNC(n) asm volatile("s_wait_asynccnt %0" :: "n"(n) : "memory")
#endif

// Low 32 bits of a generic pointer to __shared__ are the wave's LDS byte
// offset (flat LDS addr = {SHARED_BASE[63:32], offset[31:0]} per aperture map).
__device__ __forceinline__ unsigned lds_off(const void* p) {
  return (unsigned)(uintptr_t)p;
}

// One workgroup (128 threads, 4 waves) processes MPW=16 (b,c) maps.
// Pipeline: async-load map m+1 raw tiles (new&old) global->LDS while reducing
// map m; then project u/v with fp32 WMMA 16x16x4 and write one deterministic
// partial per block.
template<int IN, int NCH>
__global__ __launch_bounds__(128)
void pod_scale_kernel(const float* __restrict__ fnew,
                      const float* __restrict__ fold,
                      float* __restrict__ partial) {
  constexpr int TP   = IN + 1;                 // odd LDS pitch: conflict-free
  constexpr int UP   = NCH * 4;                // u/v length padded to K mult of 4
  constexpr int VEC  = (IN % 4 == 0) ? 4 : 2;  // rows never split a vector
  constexpr int NCHK = (IN * IN) / VEC;        // vectors per map
  constexpr int CPT  = (NCHK + 127) / 128;     // vectors per thread (uniform)

  __shared__ float TN[2][IN * TP];             // raw 'new' tiles (dbl-buffered)
  __shared__ float TO[2][IN * TP];             // raw 'old' tiles
  __shared__ float Ubuf[MPW * UP];
  __shared__ float Vbuf[MPW * UP];
  __shared__ float Acoef[IN];
  __shared__ float red[128];

  const int tid  = threadIdx.x;
  const int lane = tid & 31;
  const int wave = tid >> 5;
  const int mapBase = blockIdx.x * MPW;

  auto issue = [&](int m, int buf) {
    const float* pn = fnew + (size_t)(mapBase + m) * (size_t)(IN * IN);
    const float* po = fold + (size_t)(mapBase + m) * (size_t)(IN * IN);
    #pragma unroll
    for (int t = 0; t < CPT; ++t) {
      int ch = tid + t * 128;
      if (ch >= NCHK) ch = NCHK - 1;           // duplicate write: benign, keeps
      int idx = ch * VEC;                      // per-thread issue count uniform
      int row = idx / IN, col = idx % IN;
      unsigned ln = lds_off(&TN[buf][row * TP + col]);
      unsigned lo = lds_off(&TO[buf][row * TP + col]);
      unsigned go = (unsigned)idx * 4u;        // byte offset within map
      if (VEC == 4) {
        asm volatile("global_load_async_to_lds_b128 %0, %1, %2"
                     :: "v"(ln), "v"(go), "s"(pn) : "memory");
        asm volatile("global_load_async_to_lds_b128 %0, %1, %2"
                     :: "v"(lo), "v"(go), "s"(po) : "memory");
      } else {
        asm volatile("global_load_async_to_lds_b64 %0, %1, %2"
                     :: "v"(ln), "v"(go), "s"(pn) : "memory");
        asm volatile("global_load_async_to_lds_b64 %0, %1, %2"
                     :: "v"(lo), "v"(go), "s"(po) : "memory");
      }
    }
  };

  issue(0, 0);   // get HBM going immediately

  // A[i] = (1/32) * sum_{r<32} wt(r,i), built while the first loads fly
  for (int i = tid; i < IN; i += 128) Acoef[i] = 0.0f;
  __syncthreads();
  if (tid < 32) {
    float x = (tid + 0.5f) * ((float)IN / 64.0f) - 0.5f;
    int i0 = (int)floorf(x);
    float fr = x - (float)i0;
    if (i0 < 0)            atomicAdd(&Acoef[0],      1.0f / 32.0f);
    else if (i0 >= IN - 1) atomicAdd(&Acoef[IN - 1], 1.0f / 32.0f);
    else {
      atomicAdd(&Acoef[i0],     (1.0f - fr) / 32.0f);
      atomicAdd(&Acoef[i0 + 1],  fr         / 32.0f);
    }
  }
  __syncthreads();

  for (int m = 0; m < MPW; ++m) {
    if (m + 1 < MPW) {                 // overlap next map's HBM traffic
      issue(m + 1, (m + 1) & 1);
      WAIT_ASYNC(2 * CPT);             // in-order: map m's loads have landed
    } else {
      WAIT_ASYNC(0);
    }
    __syncthreads();

    const float* tn = TN[m & 1];
    const float* to_ = TO[m & 1];
    const int half = tid >> 6;         // waves 0,1: u (col sums); 2,3: v (rows)
    const int j    = tid & 63;
    if (j < UP) {
      float s = 0.0f;
      if (j < IN) {
        if (half == 0) {
          for (int i = 0; i < IN; ++i)
            s += Acoef[i] * (tn[i * TP + j] - to_[i * TP + j]);
        } else {
          for (int i = 0; i < IN; ++i)
            s += Acoef[i] * (tn[j * TP + i] - to_[j * TP + i]);
        }
      }
      (half ? Vbuf : Ubuf)[m * UP + j] = s;   // zero-padded K tail
    }
    __syncthreads();
  }

  // ---- WMMA fp32 16x16x4: Left(32x16) = W @ U^T, Right(32x16) = W @ V^T.
  // Each W row has <=2 taps; precompute per-lane taps branchlessly so the
  // WMMA loop is pure selects + ds_load_b64 + v_wmma (no exec-mask churn).
  const float* SRC = (wave < 2) ? Ubuf : Vbuf;
  const int rhalf  = wave & 1;
  const int mrow   = lane & 15;        // A: M / B: N(map) / D: N
  const int khi    = lane >> 4;        // lane-half K offset per ISA layout
  const int r      = rhalf * 16 + mrow;

  float x  = (r + 0.5f) * ((float)IN / 64.0f) - 0.5f;
  int   i0 = (int)floorf(x);
  float fr = x - (float)i0;
  const bool lo = (i0 < 0), hi = (i0 >= IN - 1);
  const int   t0 = lo ? 0 : (hi ? IN - 1 : i0);
  const float w0 = (lo || hi) ? 1.0f : 1.0f - fr;
  const float w1 = (lo || hi) ? 0.0f : fr;
  const int   t1 = t0 + 1;

  v8f acc = {};
  #pragma unroll
  for (int c = 0; c < NCH; ++c) {
    const int kb = 4 * c + 2 * khi;    // A/B: K = vgpr + 2*(lane/16)
    v2f a, b;
    a.x = (kb     == t0 ? w0 : 0.0f) + (kb     == t1 ? w1 : 0.0f);
    a.y = (kb + 1 == t0 ? w0 : 0.0f) + (kb + 1 == t1 ? w1 : 0.0f);
    b.x = SRC[mrow * UP + kb];
    b.y = SRC[mrow * UP + kb + 1];
    acc = __builtin_amdgcn_wmma_f32_16x16x4_f32(
        /*neg_a=*/false, a, /*neg_b=*/false, b,
        /*c_mod=*/(short)0, acc, /*reuse_a=*/false, /*reuse_b=*/false);
  }

  float s = 0.0f;
  #pragma unroll
  for (int g = 0; g < 8; ++g) s += acc[g] * acc[g];
  red[tid] = s;
  __syncthreads();
  if (tid == 0) {
    float t = 0.0f;
    for (int k = 0; k < 128; ++k) t += red[k];
    partial[blockIdx.x] = t;           // deterministic slot write, no atomics
  }
}

__global__ __launch_bounds__(256)
void pod_finalize_kernel(const float* __restrict__ partials,
                         float* __restrict__ out, int nblk) {
  __shared__ float red[256];
  float total = 0.0f;
  for (int s = 0; s < 3; ++s) {
    const float* p = partials + s * nblk;
    float t = 0.0f;
    for (int k = threadIdx.x; k < nblk; k += 256) t += p[k];
    red[threadIdx.x] = t;
    __syncthreads();
    for (int off = 128; off > 0; off >>= 1) {
      if (threadIdx.x < (unsigned)off) red[threadIdx.x] += red[threadIdx.x + off];
      __syncthreads();
    }
    if (threadIdx.x == 0) total += sqrtf(red[0]);
    __syncthreads();
  }
  if (threadIdx.x == 0) out[0] = (1e-6f + total) / 3.0f;
}

extern "C" void kernel_launch(void* const* d_in, const int* in_sizes, int n_in,
                              void* d_out, int out_size, void* d_ws, size_t ws_size,
                              hipStream_t stream) {
  (void)in_sizes; (void)n_in; (void)out_size; (void)ws_size;
  // input order: new_f0, new_f1, new_f2, old_f0, old_f1, old_f2, out_h, out_w
  const float* nf0 = (const float*)d_in[0];
  const float* nf1 = (const float*)d_in[1];
  const float* nf2 = (const float*)d_in[2];
  const float* of0 = (const float*)d_in[3];
  const float* of1 = (const float*)d_in[4];
  const float* of2 = (const float*)d_in[5];
  float* ws  = (float*)d_ws;          // 3 * NBLK floats of partials (6 KB)
  float* out = (float*)d_out;

  pod_scale_kernel<56, 14><<<NBLK, 128, 0, stream>>>(nf0, of0, ws + 0 * NBLK);
  pod_scale_kernel<28,  7><<<NBLK, 128, 0, stream>>>(nf1, of1, ws + 1 * NBLK);
  pod_scale_kernel<14,  4><<<NBLK, 128, 0, stream>>>(nf2, of2, ws + 2 * NBLK);
  pod_finalize_kernel<<<1, 256, 0, stream>>>(ws, out, NBLK);
}